// PyramidRoiAlign_36807869727172
// MI455X (gfx1250) — compile-verified
//
#include <hip/hip_runtime.h>

typedef float v4f __attribute__((ext_vector_type(4)));
typedef int   v4i __attribute__((ext_vector_type(4)));

#define CROP   14
#define MAXBS  64
#define NLVL   4
#define BATCH  4
#define NBOX   128
#define CHN    256
#define PIX    (CROP * CROP)            // 196
#define SLOTE  (PIX * CHN)              // 50176 floats per output slot
#define NSLOT  (NLVL * BATCH * MAXBS)   // 1024 slots
#define IMGM1  1023.0f                  // images are 1024x1024

// CDNA5 async global->LDS path (ASYNCcnt). Guarded: falls back to direct
// b128 gathers + global_prefetch_b8 if the builtin is absent on this toolchain.
#if defined(__gfx1250__) && __has_builtin(__builtin_amdgcn_global_load_async_to_lds_b128)
#define ASYNC_PIPE 1
#else
#define ASYNC_PIPE 0
#endif

#if ASYNC_PIPE
__device__ __forceinline__ void async_ld16(const v4f* gsrc, v4f* ldst) {
  // signature (probe-confirmed): (int4 AS1*, int4 AS3*, imm offset, imm cpol)
  __builtin_amdgcn_global_load_async_to_lds_b128(
      (__attribute__((address_space(1))) v4i*)gsrc,
      (__attribute__((address_space(3))) v4i*)ldst, 0, 0);
}
#if __has_builtin(__builtin_amdgcn_s_wait_asynccnt)
#define WAIT_ASYNC(n) __builtin_amdgcn_s_wait_asynccnt(n)
#else
#define WAIT_ASYNC(n) asm volatile("s_wait_asynccnt %0" :: "i"(n) : "memory")
#endif
#endif

#if defined(__gfx1250__) && __has_builtin(__builtin_amdgcn_ballot_w32)
#define BALLOT32(p) __builtin_amdgcn_ballot_w32(p)
#else
#define BALLOT32(p) ((unsigned)__ballot(p))
#endif

struct PixG {
  const v4f *p00, *p01, *p10, *p11;
  float wx, wy, valid;
};

__device__ __forceinline__ PixG make_geom(const float* __restrict__ fm, int S, float Sm1, int b,
                                          float ybase, float yspan, float xbase, float xspan,
                                          int pix, int cth) {
  int i = pix / CROP;
  int j = pix - i * CROP;
  float ty = (float)i / 13.0f;            // matches jnp.arange(14)/(14-1)
  float tx = (float)j / 13.0f;
  float ys = ybase + ty * yspan;
  float xs = xbase + tx * xspan;
  float y0f = floorf(ys), x0f = floorf(xs);
  PixG g;
  g.wy = ys - y0f;
  g.wx = xs - x0f;
  int y0 = (int)y0f; y0 = y0 < 0 ? 0 : (y0 > S - 1 ? S - 1 : y0);
  int y1 = (y0 + 1 > S - 1) ? S - 1 : y0 + 1;
  int x0 = (int)x0f; x0 = x0 < 0 ? 0 : (x0 > S - 1 ? S - 1 : x0);
  int x1 = (x0 + 1 > S - 1) ? S - 1 : x0 + 1;
  g.valid = (ys >= 0.0f && ys <= Sm1 && xs >= 0.0f && xs <= Sm1) ? 1.0f : 0.0f;
  const v4f* F = (const v4f*)fm;
  size_t r0 = ((size_t)b * S + y0) * (size_t)S;
  size_t r1 = ((size_t)b * S + y1) * (size_t)S;
  g.p00 = F + (r0 + x0) * (CHN / 4) + cth;
  g.p01 = F + (r0 + x1) * (CHN / 4) + cth;
  g.p10 = F + (r1 + x0) * (CHN / 4) + cth;
  g.p11 = F + (r1 + x1) * (CHN / 4) + cth;
  return g;
}

__device__ __forceinline__ v4f blend(v4f g00, v4f g01, v4f g10, v4f g11, const PixG& g) {
  float omwx = 1.0f - g.wx, omwy = 1.0f - g.wy;
  v4f top = g00 * omwx + g01 * g.wx;     // same op order as reference
  v4f bot = g10 * omwx + g11 * g.wx;
  v4f o = top * omwy + bot * g.wy;
  return o * g.valid;
}

// grid = NSLOT blocks, blockIdx.x = (lvl*BATCH + b)*MAXBS + slot; 256 threads.
__global__ __launch_bounds__(256) void roi_align_kernel(
    const float* __restrict__ f0, const float* __restrict__ f1,
    const float* __restrict__ f2, const float* __restrict__ f3,
    const float* __restrict__ dist, float* __restrict__ out) {
  const int bx   = blockIdx.x;
  const int tid  = threadIdx.x;
  const int lvl  = bx >> 8;
  const int b    = (bx >> 6) & 3;
  const int slot = bx & 63;

  // Wave32 ballot-based rank scan: threads 0..127 (4 waves) each test one box;
  // the slot-th set bit across the 4 per-wave masks is the source box index.
  __shared__ unsigned smask[4];
  {
    bool pred = false;
    if (tid < NBOX) {
      float idv = dist[((size_t)b * NBOX + tid) * 7];
      pred = ((int)idv == lvl);
    }
    unsigned m = BALLOT32(pred);
    if (tid < NBOX && (tid & 31) == 0) smask[tid >> 5] = m;
  }
  __syncthreads();

  int nfound = -1;
  {
    int rem = slot;
    #pragma unroll
    for (int w2 = 0; w2 < 4; ++w2) {
      unsigned m = smask[w2];
      int c = __popc(m);
      if (nfound < 0) {
        if (rem < c) {
          unsigned mm = m;
          for (int k = 0; k < rem; ++k) mm &= mm - 1;  // drop lowest rem bits
          nfound = w2 * 32 + __builtin_ctz(mm);
          rem = 0;
        } else {
          rem -= c;
        }
      }
    }
  }

  // roi_boxes output: [B, NLVL*MAXBS, 6] appended after the fmap block.
  if (tid == 0) {
    float* p = out + (size_t)NSLOT * SLOTE +
               ((size_t)b * (NLVL * MAXBS) + (size_t)lvl * MAXBS + slot) * 6;
    if (nfound >= 0) {
      const float* d = dist + ((size_t)b * NBOX + nfound) * 7;
      #pragma unroll
      for (int k = 0; k < 6; ++k) p[k] = d[1 + k];
    } else {
      #pragma unroll
      for (int k = 0; k < 6; ++k) p[k] = 0.0f;
    }
  }

  v4f* ob = (v4f*)out + (size_t)bx * (SLOTE / 4);

  if (nfound < 0) {                      // empty slot: stream NT zeros
    v4f z = {0.0f, 0.0f, 0.0f, 0.0f};
    #pragma unroll 1
    for (int it = 0; it < SLOTE / (256 * 4); ++it)   // 49 iters
      __builtin_nontemporal_store(z, ob + (size_t)it * 256 + tid);
    return;
  }

  const int S = 256 >> lvl;
  const float Sm1 = (float)(S - 1);
  const float* fm = (lvl == 0) ? f0 : (lvl == 1) ? f1 : (lvl == 2) ? f2 : f3;
  const float* d = dist + ((size_t)b * NBOX + nfound) * 7;
  const float cx = d[1], cy = d[2], w = d[3], h = d[4];
  const float ny1 = (cy - h * 0.5f) / IMGM1;   // true divides, as in reference
  const float nx1 = (cx - w * 0.5f) / IMGM1;
  const float ny2 = (cy + h * 0.5f) / IMGM1;
  const float nx2 = (cx + w * 0.5f) / IMGM1;
  const float ybase = ny1 * Sm1, yspan = (ny2 - ny1) * Sm1;
  const float xbase = nx1 * Sm1, xspan = (nx2 - nx1) * Sm1;

  const int cth  = tid & 63;     // 64 threads cover C=256 as float4
  const int pgrp = tid >> 6;     // 4 pixels in flight per iteration

#if ASYNC_PIPE
  // Double-buffered async global->LDS pipeline (ASYNCcnt; in-order completion).
  __shared__ v4f stage[2][4][256];
  int cur = 0;
  PixG g = make_geom(fm, S, Sm1, b, ybase, yspan, xbase, xspan, pgrp, cth);
  async_ld16(g.p00, &stage[0][0][tid]);
  async_ld16(g.p01, &stage[0][1][tid]);
  async_ld16(g.p10, &stage[0][2][tid]);
  async_ld16(g.p11, &stage[0][3][tid]);
  #pragma unroll 1
  for (int it = 0; it < 49; ++it) {
    PixG gn = g;
    if (it < 48) {
      gn = make_geom(fm, S, Sm1, b, ybase, yspan, xbase, xspan, (it + 1) * 4 + pgrp, cth);
      const int nb = cur ^ 1;
      async_ld16(gn.p00, &stage[nb][0][tid]);
      async_ld16(gn.p01, &stage[nb][1][tid]);
      async_ld16(gn.p10, &stage[nb][2][tid]);
      async_ld16(gn.p11, &stage[nb][3][tid]);
      WAIT_ASYNC(4);                     // current buffer complete (in-order)
    } else {
      WAIT_ASYNC(0);
    }
    v4f a0 = stage[cur][0][tid];
    v4f a1 = stage[cur][1][tid];
    v4f a2 = stage[cur][2][tid];
    v4f a3 = stage[cur][3][tid];
    v4f o = blend(a0, a1, a2, a3, g);
    __builtin_nontemporal_store(o, ob + (size_t)(it * 4 + pgrp) * (CHN / 4) + cth);
    g = gn; cur ^= 1;
  }
#else
  // Fallback: direct b128 gathers, software-pipelined geometry + prefetch.
  PixG g = make_geom(fm, S, Sm1, b, ybase, yspan, xbase, xspan, pgrp, cth);
  #pragma unroll 1
  for (int it = 0; it < 49; ++it) {
    PixG gn = g;
    if (it < 48) {
      gn = make_geom(fm, S, Sm1, b, ybase, yspan, xbase, xspan, (it + 1) * 4 + pgrp, cth);
      __builtin_prefetch((const void*)gn.p00, 0, 0);   // global_prefetch_b8
      __builtin_prefetch((const void*)gn.p01, 0, 0);
      __builtin_prefetch((const void*)gn.p10, 0, 0);
      __builtin_prefetch((const void*)gn.p11, 0, 0);
    }
    v4f a0 = *g.p00;
    v4f a1 = *g.p01;
    v4f a2 = *g.p10;
    v4f a3 = *g.p11;
    v4f o = blend(a0, a1, a2, a3, g);
    __builtin_nontemporal_store(o, ob + (size_t)(it * 4 + pgrp) * (CHN / 4) + cth);
    g = gn;
  }
#endif
}

extern "C" void kernel_launch(void* const* d_in, const int* in_sizes, int n_in,
                              void* d_out, int out_size, void* d_ws, size_t ws_size,
                              hipStream_t stream) {
  const float* f0   = (const float*)d_in[0];
  const float* f1   = (const float*)d_in[1];
  const float* f2   = (const float*)d_in[2];
  const float* f3   = (const float*)d_in[3];
  const float* dist = (const float*)d_in[4];
  // d_in[5] (images) only defines Hi=Wi=1024, hardcoded above.
  float* out = (float*)d_out;
  (void)in_sizes; (void)n_in; (void)out_size; (void)d_ws; (void)ws_size;

  roi_align_kernel<<<dim3(NSLOT), dim3(256), 0, stream>>>(f0, f1, f2, f3, dist, out);
}